// STLayer_38878043963794
// MI455X (gfx1250) — compile-verified
//
#include <hip/hip_runtime.h>

typedef __attribute__((ext_vector_type(2))) float v2f;
typedef __attribute__((ext_vector_type(8))) float v8f;

#define HD  256      // hidden size H
#define NB  8        // batch B
#define FPB 64       // facts per block in scatter kernel

// ---------------------------------------------------------------------------
// Kernel 0: zero the output accumulator (16 MB), float4-vectorized.
// ---------------------------------------------------------------------------
__global__ __launch_bounds__(256)
void zero_kernel(float4* __restrict__ out, int n4) {
    int i = blockIdx.x * blockDim.x + threadIdx.x;
    if (i < n4) out[i] = make_float4(0.f, 0.f, 0.f, 0.f);
}

// ---------------------------------------------------------------------------
// Kernel 1: P = rel_features @ W^T + b   (NR x H, fp32, exact)
// One wave per 16x16 output tile, V_WMMA_F32_16X16X4_F32 chained over K=256.
//
// A (16x4 fp32): lanes 0-15 hold rows M=0..15 with K={0,1}; lanes 16-31 hold
// the same rows with K={2,3}  -> per-lane float2 load at k0 + 2*(lane>>4).
// B (4x16 fp32): mirrored layout; B[k][n] = W[n][k], so lane l loads the
// contiguous pair W[n0+l][k0 + 2*(lane>>4) + {0,1}].
// C/D (16x16 fp32): acc[v] holds row m0 + v + 8*(lane>>4), col n0 + (lane&15).
// ---------------------------------------------------------------------------
__global__ __launch_bounds__(32)
void relproj_wmma_kernel(const float* __restrict__ rel,   // [NR, H]
                         const float* __restrict__ W,     // [H, H]
                         const float* __restrict__ bias,  // [H]
                         float* __restrict__ P,           // [NR, H]
                         int NR) {
    const int lane = threadIdx.x;     // 0..31 (wave32)
    const int half = lane >> 4;       // K-pair selector
    const int l    = lane & 15;
    const int m0   = blockIdx.x * 16;
    const int n0   = blockIdx.y * 16;

    // Clamp A row so EXEC stays all-ones for WMMA even on a ragged last tile.
    int arow = m0 + l;
    if (arow >= NR) arow = NR - 1;

    const float* Ap = rel + (size_t)arow * HD + 2 * half;
    const float* Bp = W   + (size_t)(n0 + l) * HD + 2 * half;

    v8f acc = {};
#pragma unroll 8
    for (int k0 = 0; k0 < HD; k0 += 4) {
        v2f a, b;
        a.x = Ap[k0 + 0]; a.y = Ap[k0 + 1];
        b.x = Bp[k0 + 0]; b.y = Bp[k0 + 1];
        // D = A*B + C  (8 args: neg_a, A, neg_b, B, c_mod, C, reuse_a, reuse_b)
        acc = __builtin_amdgcn_wmma_f32_16x16x4_f32(
                  false, a, false, b, (short)0, acc, false, false);
    }

    const float bn  = bias[n0 + l];
    float* Pp = P + (size_t)(m0 + 8 * half) * HD + n0 + l;

    if (m0 + 16 <= NR) {
        // Full tile (the only case when NR % 16 == 0): uniform scalar branch,
        // no per-element exec-mask churn.
#pragma unroll
        for (int v = 0; v < 8; ++v)
            Pp[(size_t)v * HD] = acc[v] + bn;
    } else {
        // Ragged last tile: per-row guard.
#pragma unroll
        for (int v = 0; v < 8; ++v) {
            int row = m0 + v + 8 * half;
            if (row < NR) Pp[(size_t)v * HD] = acc[v] + bn;
        }
    }
}

// ---------------------------------------------------------------------------
// Kernel 2: per-fact gather/scale/relu/scatter-add.
//   out[tails[f]] += relu(P[rels[f]] * instr[ids[f]]) * dist[heads[f]]
// 256 threads = one H-row; FPB facts per block. instruction (8 KB) in LDS.
// Atomics skipped where relu produced exact zero (~50% of elements).
// ---------------------------------------------------------------------------
__global__ __launch_bounds__(256)
void facts_kernel(const float* __restrict__ P,      // [NR, H]
                  const float* __restrict__ instr,  // [NB, H]
                  const float* __restrict__ dist,   // [B*M] flattened
                  const int*  __restrict__ rels,
                  const int*  __restrict__ ids,
                  const int*  __restrict__ heads,
                  const int*  __restrict__ tails,
                  float* __restrict__ out,          // [B*M, H]
                  int nf) {
    __shared__ float instr_s[NB * HD];
    for (int i = threadIdx.x; i < NB * HD; i += blockDim.x)
        instr_s[i] = instr[i];
    __syncthreads();

    const int h    = threadIdx.x;          // 0..255 -> column, coalesced
    const int f0   = blockIdx.x * FPB;
    const int fend = (f0 + FPB < nf) ? (f0 + FPB) : nf;

    for (int f = f0; f < fend; ++f) {
        // Uniform per-iteration scalars -> compiler lowers to s_load.
        const int r  = rels[f];
        const int bi = ids[f];
        const int hd = heads[f];
        const int tl = tails[f];

        if (f + 1 < fend) {
            // Warm the next fact's P row (L2-resident; hides gather latency).
            __builtin_prefetch(&P[(size_t)rels[f + 1] * HD + h], 0, 0);
        }

        const float prior = dist[hd];
        float v = P[(size_t)r * HD + h] * instr_s[bi * HD + h];
        v = fmaxf(v, 0.0f) * prior;
        if (v != 0.0f)
            atomicAdd(&out[(size_t)tl * HD + h], v);
    }
}

// ---------------------------------------------------------------------------
extern "C" void kernel_launch(void* const* d_in, const int* in_sizes, int n_in,
                              void* d_out, int out_size, void* d_ws, size_t ws_size,
                              hipStream_t stream) {
    // setup_inputs() order:
    // 0 input_vector [B,M,H] (unused by the math), 1 curr_dist [B,M],
    // 2 instruction [B,H], 3 rel_features [NR,H], 4 W [H,H], 5 b [H],
    // 6 batch_heads [NF], 7 batch_rels [NF], 8 batch_tails [NF], 9 batch_ids [NF]
    const float* curr_dist    = (const float*)d_in[1];
    const float* instruction  = (const float*)d_in[2];
    const float* rel_features = (const float*)d_in[3];
    const float* W            = (const float*)d_in[4];
    const float* bias         = (const float*)d_in[5];
    const int*   batch_heads  = (const int*)d_in[6];
    const int*   batch_rels   = (const int*)d_in[7];
    const int*   batch_tails  = (const int*)d_in[8];
    const int*   batch_ids    = (const int*)d_in[9];

    const int NR = in_sizes[3] / HD;   // 2000
    const int nf = in_sizes[6];        // 500000

    float* P   = (float*)d_ws;         // NR*H*4 = 2 MB scratch
    float* out = (float*)d_out;        // B*M*H fp32

    // 1) zero accumulator
    const int n4 = out_size / 4;
    zero_kernel<<<(n4 + 255) / 256, 256, 0, stream>>>((float4*)out, n4);

    // 2) P = rel_features @ W^T + b   (fp32 WMMA)
    dim3 g1((NR + 15) / 16, HD / 16);
    relproj_wmma_kernel<<<g1, 32, 0, stream>>>(rel_features, W, bias, P, NR);

    // 3) gather / gate / relu / prior-scale / scatter-add
    facts_kernel<<<(nf + FPB - 1) / FPB, 256, 0, stream>>>(
        P, instruction, curr_dist,
        batch_rels, batch_ids, batch_heads, batch_tails, out, nf);
}